// MobiusGRU_52432960749611
// MI455X (gfx1250) — compile-verified
//
#include <hip/hip_runtime.h>
#include <hip/hip_bf16.h>

#define T_ 128
#define B_ 64
#define D_ 1024
#define H_ 1024
#define EPSF 1e-15f
#define MAXNF (1.0f - 1e-5f)
#define KSLICES 4
#define PS_ZR (2ull * B_ * 2 * H_)   // partial-slice stride for h@[wz|wr]
#define PS_HH (2ull * B_ * H_)       // partial-slice stride for rh@wh

typedef float v2f __attribute__((ext_vector_type(2)));
typedef float v4f __attribute__((ext_vector_type(4)));
typedef float v8f __attribute__((ext_vector_type(8)));

// ---------------------------------------------------------------------------
// block-wide reductions (blockDim.x == 256 => 8 waves of 32 on gfx1250)
// ---------------------------------------------------------------------------
__device__ __forceinline__ float block_sum(float v, float* s16) {
#pragma unroll
  for (int o = 16; o; o >>= 1) v += __shfl_xor(v, o, 32);
  __syncthreads();
  if ((threadIdx.x & 31) == 0) s16[threadIdx.x >> 5] = v;
  __syncthreads();
  float r = 0.f;
#pragma unroll
  for (int i = 0; i < 8; ++i) r += s16[i];
  return r;
}

__device__ __forceinline__ void block_sum2(float& a, float& b, float* s16) {
#pragma unroll
  for (int o = 16; o; o >>= 1) {
    a += __shfl_xor(a, o, 32);
    b += __shfl_xor(b, o, 32);
  }
  __syncthreads();
  if ((threadIdx.x & 31) == 0) {
    int w = threadIdx.x >> 5;
    s16[w] = a;
    s16[8 + w] = b;
  }
  __syncthreads();
  a = 0.f; b = 0.f;
#pragma unroll
  for (int i = 0; i < 8; ++i) { a += s16[i]; b += s16[8 + i]; }
}

__device__ __forceinline__ float dot4(v4f a, v4f b) {
  return a[0] * b[0] + a[1] * b[1] + a[2] * b[2] + a[3] * b[3];
}
__device__ __forceinline__ float sum4(v4f a) { return a[0] + a[1] + a[2] + a[3]; }

// ---------------------------------------------------------------------------
// scalar hyperbolic helpers
// ---------------------------------------------------------------------------
__device__ __forceinline__ float atanh_clip(float x) {
  x = fminf(fmaxf(x, -MAXNF), MAXNF);
  return atanhf(x);
}

// s such that _mob_rescale(mx, x) == s * mx, given raw sum(mx^2), sum(x^2)
__device__ __forceinline__ float rescale_scale(float m2raw, float x2raw) {
  float xn  = sqrtf(x2raw + EPSF);
  float mxn = sqrtf(m2raw + EPSF);
  float s = tanhf(mxn / xn * atanh_clip(xn)) / mxn;
  float nn = sqrtf(s * s * m2raw + EPSF);   // proj norm of result
  if (nn > MAXNF) s *= MAXNF / nn;
  return s;
}

// mob_add(x,y) == ca*x + cb*y ; r2 = raw sum of squares of the result
__device__ __forceinline__ void mob_add_coef(float xy, float x2, float y2,
                                             float& ca, float& cb, float& r2) {
  float a   = 1.f + 2.f * xy + y2;
  float bb  = 1.f - x2;
  float den = 1.f + 2.f * xy + x2 * y2 + EPSF;
  float inv = 1.f / den;
  float num2 = a * a * x2 + bb * bb * y2 + 2.f * a * bb * xy;
  float nn = sqrtf(num2 * inv * inv + EPSF);  // proj norm
  float sc = inv;
  if (nn > MAXNF) sc *= MAXNF / nn;
  ca = sc * a;
  cb = sc * bb;
  r2 = ca * ca * x2 + cb * cb * y2 + 2.f * ca * cb * xy;
}

// ---------------------------------------------------------------------------
// fp32 WMMA GEMM, 2-deep software pipeline.  C[M,N] (+k-sliced) = A[M,K]@W[K,N],
// W passed TRANSPOSED (Wt is N x K).  Wave tile 32x64: 8 accumulators,
// 8 v_wmma_f32_16x16x4_f32 per 6 v2f loads; fragments for k+4 and k+8 are in
// flight while the WMMAs for k execute, giving the wait-inserter ~12 loads of
// slack (partial s_wait_loadcnt, full overlap of VMEM with the matrix pipe).
// grid: x = N/512, y = (M/32)*kslices, z = operand set.
// Each k-slice writes its own partial C at C + ks*cks (no atomics).
// ---------------------------------------------------------------------------
struct G4 { const float* A[4]; const float* Wt[4]; float* C[4]; };

struct Frag { v2f a0, a1, b0, b1, b2, b3; };

#define WMMA_F32(acc, a, b) \
  acc = __builtin_amdgcn_wmma_f32_16x16x4_f32(false, a, false, b, (short)0, acc, false, false)

__global__ __launch_bounds__(256) void k_wmma_gemm(G4 g, int M, int N, int K,
                                                   int kslices, unsigned long long cks) {
  const float* __restrict__ A  = g.A[blockIdx.z];
  const float* __restrict__ Wt = g.Wt[blockIdx.z];
  const int mt = blockIdx.y / kslices;
  const int ks = blockIdx.y % kslices;
  float* __restrict__ C = g.C[blockIdx.z] + (size_t)ks * cks;
  const int Ks = K / kslices;
  const int k0 = ks * Ks;

  const int lane = threadIdx.x & 31;
  const int wave = threadIdx.x >> 5;
  const int m0 = mt << 5;
  const int n0 = (blockIdx.x << 9) + (wave << 6);
  const int ar   = lane & 15;
  const int koff = (lane >> 4) << 1;

  const float* ap0 = A + (size_t)(m0 + ar) * K + k0 + koff;
  const float* ap1 = ap0 + (size_t)16 * K;
  const float* w0  = Wt + (size_t)(n0 + ar) * K + k0 + koff;
  const float* w1  = w0 + (size_t)16 * K;
  const float* w2  = w0 + (size_t)32 * K;
  const float* w3  = w0 + (size_t)48 * K;

  v8f acc00 = {}, acc01 = {}, acc02 = {}, acc03 = {};
  v8f acc10 = {}, acc11 = {}, acc12 = {}, acc13 = {};

#define LOAD_FRAG(f, k)                       \
  do {                                        \
    (f).a0 = *(const v2f*)(ap0 + (k));        \
    (f).a1 = *(const v2f*)(ap1 + (k));        \
    (f).b0 = *(const v2f*)(w0 + (k));         \
    (f).b1 = *(const v2f*)(w1 + (k));         \
    (f).b2 = *(const v2f*)(w2 + (k));         \
    (f).b3 = *(const v2f*)(w3 + (k));         \
  } while (0)

#define DO_WMMA(f)                                              \
  do {                                                          \
    WMMA_F32(acc00, (f).a0, (f).b0); WMMA_F32(acc01, (f).a0, (f).b1); \
    WMMA_F32(acc02, (f).a0, (f).b2); WMMA_F32(acc03, (f).a0, (f).b3); \
    WMMA_F32(acc10, (f).a1, (f).b0); WMMA_F32(acc11, (f).a1, (f).b1); \
    WMMA_F32(acc12, (f).a1, (f).b2); WMMA_F32(acc13, (f).a1, (f).b3); \
  } while (0)

  Frag cur, nxt;
  LOAD_FRAG(cur, 0);
  LOAD_FRAG(nxt, 4);
  // Ks is a multiple of 8 (256 or 1024)
  for (int k = 8; k < Ks; k += 8) {
    Frag f1, f2;
    LOAD_FRAG(f1, k);       // two iterations ahead of cur's consumption
    DO_WMMA(cur);
    LOAD_FRAG(f2, k + 4);
    DO_WMMA(nxt);
    cur = f1;
    nxt = f2;
  }
  DO_WMMA(cur);
  DO_WMMA(nxt);
#undef LOAD_FRAG
#undef DO_WMMA

  const int cm  = m0 + ((lane >> 4) << 3);
  const int cn0 = n0 + ar;
#pragma unroll
  for (int i = 0; i < 8; ++i) {
    size_t rb = (size_t)(cm + i) * N;
    C[rb + cn0     ] = acc00[i];
    C[rb + cn0 + 16] = acc01[i];
    C[rb + cn0 + 32] = acc02[i];
    C[rb + cn0 + 48] = acc03[i];
    rb += (size_t)16 * N;
    C[rb + cn0     ] = acc10[i];
    C[rb + cn0 + 16] = acc11[i];
    C[rb + cn0 + 32] = acc12[i];
    C[rb + cn0 + 48] = acc13[i];
  }
}

// ---------------------------------------------------------------------------
// tiled 1024x1024 transpose: out[n*1024+k] = in[k*1024+n]
// ---------------------------------------------------------------------------
__global__ __launch_bounds__(256) void k_transpose_1k(const float* __restrict__ in,
                                                      float* __restrict__ out) {
  __shared__ float tile[32][33];
  const int bx = blockIdx.x << 5, by = blockIdx.y << 5;
  const int tx = threadIdx.x & 31, ty = threadIdx.x >> 5;  // 32x8
#pragma unroll
  for (int j = 0; j < 32; j += 8)
    tile[ty + j][tx] = in[(size_t)(by + ty + j) * 1024 + bx + tx];
  __syncthreads();
#pragma unroll
  for (int j = 0; j < 32; j += 8)
    out[(size_t)(bx + ty + j) * 1024 + by + tx] = tile[tx][ty + j];
}

__global__ void k_zero(float* p, int n) {
  int i = blockIdx.x * 256 + threadIdx.x;
  if (i < n) p[i] = 0.f;
}

// ---------------------------------------------------------------------------
// apply _mob_rescale(x@U, x) in place on the 4 hoisted projection buffers
// ---------------------------------------------------------------------------
struct XuPtrs { float* buf[4]; };

__global__ __launch_bounds__(256) void k_xu_rescale(const float* __restrict__ x, XuPtrs xp) {
  __shared__ float sred[16];
  const size_t row = blockIdx.x;
  const int tid = threadIdx.x;
  const v4f* xr = (const v4f*)(x + row * (size_t)D_);
  v4f* ur = (v4f*)(xp.buf[blockIdx.y] + row * (size_t)H_);
  v4f xv = xr[tid];
  v4f uv = ur[tid];
  float xs = dot4(xv, xv), us = dot4(uv, uv);
  block_sum2(xs, us, sred);
  float s = rescale_scale(us, xs);
  ur[tid] = uv * s;
}

// ---------------------------------------------------------------------------
struct StepPtrs {
  const float* xz[2]; const float* xr[2];
  const float* bz[2]; const float* br[2]; const float* bh[2];
  const float* gz[2]; const float* bez[2];
  const float* gr[2]; const float* ber[2];
};

// gates: z (stored), r, rh = mob_pointwise_prod(h, r) (stored)
// grid: (B, 2 dirs), block 256, 4 consecutive floats per thread (v4f)
__global__ __launch_bounds__(256) void k_gates(const float* __restrict__ hb,
                                               const float* __restrict__ hzr,
                                               float* __restrict__ zb,
                                               float* __restrict__ rhb,
                                               StepPtrs p, int s) {
  __shared__ float sred[16];
  const int b = blockIdx.x, dir = blockIdx.y;
  const int tt = dir ? (T_ - 1 - s) : s;
  const size_t rbase = ((size_t)dir * B_ + b) * H_;
  const int tid = threadIdx.x;

  v4f h4 = ((const v4f*)(hb + rbase))[tid];
  const float h2 = block_sum(dot4(h4, h4), sred);

  v4f rv;
  for (int g = 0; g < 2; ++g) {
    const size_t zrbase = ((size_t)dir * B_ + b) * (2 * H_) + (size_t)g * H_;
    // sum the KSLICES partial GEMM results
    v4f mv = ((const v4f*)(hzr + zrbase))[tid];
#pragma unroll
    for (int ks = 1; ks < KSLICES; ++ks)
      mv += ((const v4f*)(hzr + zrbase + (size_t)ks * PS_ZR))[tid];
    v4f xv  = ((const v4f*)((g ? p.xr[dir] : p.xz[dir]) + ((size_t)tt * B_ + b) * H_))[tid];
    v4f biv = ((const v4f*)(g ? p.br[dir]  : p.bz[dir]))[tid];
    v4f gam = ((const v4f*)(g ? p.gr[dir]  : p.gz[dir]))[tid];
    v4f bet = ((const v4f*)(g ? p.ber[dir] : p.bez[dir]))[tid];

    float mm2 = block_sum(dot4(mv, mv), sred);
    // a = _mob_rescale(h@W, h)
    float s1 = rescale_scale(mm2, h2);
    float a2 = s1 * s1 * mm2;
    mv = mv * s1;
    // m = mob_add(a, xu)
    float sx2 = dot4(xv, xv), sax = dot4(mv, xv);
    block_sum2(sx2, sax, sred);
    float ca, cb, m2;
    mob_add_coef(sax, a2, sx2, ca, cb, m2);
    mv = mv * ca + xv * cb;
    // q = mob_add(m, bias)
    float sb2 = dot4(biv, biv), smb = dot4(mv, biv);
    block_sum2(sb2, smb, sred);
    float q2;
    mob_add_coef(smb, m2, sb2, ca, cb, q2);
    mv = mv * ca + biv * cb;
    // log_map_zero + layernorm + sigmoid
    float yn = sqrtf(q2 + EPSF);
    float lf = atanh_clip(yn) / yn;
    float s1sum = block_sum(sum4(mv), sred);
    float mean = lf * s1sum * (1.f / H_);
    float var  = lf * lf * q2 * (1.f / H_) - mean * mean;
    float rstd = rsqrtf(var + 1e-5f);
    v4f sig;
#pragma unroll
    for (int j = 0; j < 4; ++j) {
      float v = gam[j] * (lf * mv[j] - mean) * rstd + bet[j];
      sig[j] = 1.f / (1.f + expf(-v));
    }
    if (g == 0) ((v4f*)(zb + rbase))[tid] = sig;
    else        rv = sig;
  }
  // rh = mob_pointwise_prod(h, r) = rescale(h*r, h)
  v4f pr = h4 * rv;
  float ps = block_sum(dot4(pr, pr), sred);
  float s2 = rescale_scale(ps, h2);
  ((v4f*)(rhb + rbase))[tid] = pr * s2;
}

// update: h_tilde chain + Möbius GRU state update; writes new h and output slot
__global__ __launch_bounds__(256) void k_update(float* __restrict__ hb,
                                                const float* __restrict__ rhb,
                                                const float* __restrict__ hhb,
                                                const float* __restrict__ zbb,
                                                float* __restrict__ out,
                                                StepPtrs p, int s) {
  __shared__ float sred[16];
  const int b = blockIdx.x, dir = blockIdx.y;
  const int tt = dir ? (T_ - 1 - s) : s;
  const size_t rbase = ((size_t)dir * B_ + b) * H_;
  const int tid = threadIdx.x;

  v4f hv  = ((const v4f*)(hb + rbase))[tid];
  v4f rhv = ((const v4f*)(rhb + rbase))[tid];
  v4f zv  = ((const v4f*)(zbb + rbase))[tid];
  v4f xv  = ((const v4f*)(p.xr[dir] + ((size_t)tt * B_ + b) * H_))[tid];
  v4f bv  = ((const v4f*)(p.bh[dir]))[tid];
  // sum the KSLICES partial rh@wh results
  v4f hhv = ((const v4f*)(hhb + rbase))[tid];
#pragma unroll
  for (int ks = 1; ks < KSLICES; ++ks)
    hhv += ((const v4f*)(hhb + rbase + (size_t)ks * PS_HH))[tid];

  float sh2 = dot4(hv, hv), srh2 = dot4(rhv, rhv);
  block_sum2(sh2, srh2, sred);
  const float h2 = sh2;
  const float hh2 = block_sum(dot4(hhv, hhv), sred);
  // c = rescale(rh@wh, rh)
  float sc = rescale_scale(hh2, srh2);
  float c2 = sc * sc * hh2;
  hhv = hhv * sc;
  // m = mob_add(c, xu_r)
  float sx2 = dot4(xv, xv), scx = dot4(hhv, xv);
  block_sum2(sx2, scx, sred);
  float ca, cb, m2;
  mob_add_coef(scx, c2, sx2, ca, cb, m2);
  v4f mv = hhv * ca + xv * cb;
  // h_tilde = mob_add(m, bh)
  float sb2 = dot4(bv, bv), smb = dot4(mv, bv);
  block_sum2(sb2, smb, sred);
  float ht2;
  mob_add_coef(smb, m2, sb2, ca, cb, ht2);
  v4f htv = mv * ca + bv * cb;
  // d = mob_add(-h, h_tilde)
  float shht = block_sum(dot4(hv, htv), sred);
  float d2;
  mob_add_coef(-shht, h2, ht2, ca, cb, d2);
  v4f dv = htv * cb - hv * ca;
  // e = mob_pointwise_prod(d, z)
  v4f dzv = dv * zv;
  float sdz2 = block_sum(dot4(dzv, dzv), sred);
  float se = rescale_scale(sdz2, d2);
  float e2 = se * se * sdz2;
  // h_new = mob_add(h, e)
  float she = block_sum(dot4(hv, dzv) * se, sred);
  float hn2;
  mob_add_coef(she, h2, e2, ca, cb, hn2);
  v4f hn = hv * ca + dzv * (cb * se);
  ((v4f*)(hb + rbase))[tid] = hn;
  float* op = out + ((size_t)tt * B_ + b) * (2 * H_) + (dir ? 0 : H_);
  ((v4f*)op)[tid] = hn;
}

// final proj over concatenated 2048-wide rows (in place on d_out)
__global__ __launch_bounds__(256) void k_proj_out(float* __restrict__ out) {
  __shared__ float sred[16];
  v4f* r = (v4f*)(out + (size_t)blockIdx.x * (2 * H_));
  v4f v0 = r[threadIdx.x];
  v4f v1 = r[threadIdx.x + 256];
  float ss = block_sum(dot4(v0, v0) + dot4(v1, v1), sred);
  float n = sqrtf(ss + EPSF);
  float scl = (n > MAXNF) ? (MAXNF / n) : 1.f;
  r[threadIdx.x]       = v0 * scl;
  r[threadIdx.x + 256] = v1 * scl;
}

// ht = stack([hTn, hTp]) appended after out
__global__ void k_tail(const float* __restrict__ hb, float* __restrict__ out) {
  int i = blockIdx.x * 256 + threadIdx.x;  // v4f units, 0..32767
  const int half = B_ * H_ / 4;
  v4f v = (i < half) ? ((const v4f*)(hb + (size_t)B_ * H_))[i]   // dir 1 = 'n' first
                     : ((const v4f*)hb)[i - half];               // dir 0 = 'p'
  ((v4f*)(out + (size_t)T_ * B_ * 2 * H_))[i] = v;
}

// ---------------------------------------------------------------------------
extern "C" void kernel_launch(void* const* d_in, const int* in_sizes, int n_in,
                              void* d_out, int out_size, void* d_ws, size_t ws_size,
                              hipStream_t stream) {
  (void)in_sizes; (void)n_in; (void)out_size; (void)ws_size;
  const float* x = (const float*)d_in[0];
  const float* wz[2] = {(const float*)d_in[1],  (const float*)d_in[13]};
  const float* wr[2] = {(const float*)d_in[2],  (const float*)d_in[14]};
  const float* wh[2] = {(const float*)d_in[3],  (const float*)d_in[15]};
  const float* uz[2] = {(const float*)d_in[4],  (const float*)d_in[16]};
  const float* ur[2] = {(const float*)d_in[5],  (const float*)d_in[17]};
  StepPtrs sp;
  sp.bz[0]  = (const float*)d_in[6];  sp.bz[1]  = (const float*)d_in[18];
  sp.br[0]  = (const float*)d_in[7];  sp.br[1]  = (const float*)d_in[19];
  sp.bh[0]  = (const float*)d_in[8];  sp.bh[1]  = (const float*)d_in[20];
  sp.gz[0]  = (const float*)d_in[9];  sp.gz[1]  = (const float*)d_in[21];
  sp.bez[0] = (const float*)d_in[10]; sp.bez[1] = (const float*)d_in[22];
  sp.gr[0]  = (const float*)d_in[11]; sp.gr[1]  = (const float*)d_in[23];
  sp.ber[0] = (const float*)d_in[12]; sp.ber[1] = (const float*)d_in[24];

  float* ws = (float*)d_ws;
  const unsigned long long TBH = (unsigned long long)T_ * B_ * H_;  // 8,388,608
  const unsigned long long M1  = 1048576ull;                         // 1024*1024
  float* XZ   = ws;                        // [2][T*B][H]
  float* XR   = XZ + 2 * TBH;              // [2][T*B][H]
  float* WZRT = XR + 2 * TBH;              // [2][2048][1024] transposed, packed wz|wr
  float* WHT  = WZRT + 2 * 2 * M1;         // [2][1024][1024] transposed
  float* UT   = WHT + 2 * M1;              // [4][1024][1024] uz_p, ur_p, uz_n, ur_n
  float* HB   = UT + 4 * M1;               // [2][B][H]
  float* HZR  = HB + 2ull * B_ * H_;       // [KSLICES][2][B][2H]
  float* HH   = HZR + KSLICES * PS_ZR;     // [KSLICES][2][B][H]
  float* RH   = HH + KSLICES * PS_HH;      // [2][B][H]
  float* ZB   = RH + 2ull * B_ * H_;       // [2][B][H]

  dim3 blk(256);
  dim3 tg(32, 32);
  for (int d = 0; d < 2; ++d) {
    k_transpose_1k<<<tg, blk, 0, stream>>>(wz[d], WZRT + d * 2 * M1);
    k_transpose_1k<<<tg, blk, 0, stream>>>(wr[d], WZRT + d * 2 * M1 + M1);
    k_transpose_1k<<<tg, blk, 0, stream>>>(wh[d], WHT + d * M1);
    k_transpose_1k<<<tg, blk, 0, stream>>>(uz[d], UT + (2 * d) * M1);
    k_transpose_1k<<<tg, blk, 0, stream>>>(ur[d], UT + (2 * d + 1) * M1);
  }
  k_zero<<<(2 * B_ * H_ + 255) / 256, blk, 0, stream>>>(HB, 2 * B_ * H_);

  // hoisted input projections: x(8192x1024) @ {uz_p, ur_p, uz_n, ur_n}
  G4 gp;
  float* xc[4] = {XZ, XR, XZ + TBH, XR + TBH};
  for (int q = 0; q < 4; ++q) { gp.A[q] = x; gp.Wt[q] = UT + q * M1; gp.C[q] = xc[q]; }
  k_wmma_gemm<<<dim3(H_ / 512, (T_ * B_) / 32, 4), blk, 0, stream>>>(
      gp, T_ * B_, H_, D_, 1, 0ull);

  XuPtrs xp; for (int q = 0; q < 4; ++q) xp.buf[q] = xc[q];
  k_xu_rescale<<<dim3(T_ * B_, 4), blk, 0, stream>>>(x, xp);
  for (int d = 0; d < 2; ++d) { sp.xz[d] = xc[2 * d]; sp.xr[d] = xc[2 * d + 1]; }

  // recurrent operand sets (grid.z = direction)
  G4 gzr, gwh;
  for (int d = 0; d < 2; ++d) {
    gzr.A[d]  = HB + (size_t)d * B_ * H_;
    gzr.Wt[d] = WZRT + d * 2 * M1;
    gzr.C[d]  = HZR + (size_t)d * B_ * 2 * H_;
    gwh.A[d]  = RH + (size_t)d * B_ * H_;
    gwh.Wt[d] = WHT + d * M1;
    gwh.C[d]  = HH + (size_t)d * B_ * H_;
  }
  for (int d = 2; d < 4; ++d) {  // unused slots (grid.z == 2)
    gzr.A[d] = gzr.A[0]; gzr.Wt[d] = gzr.Wt[0]; gzr.C[d] = gzr.C[0];
    gwh.A[d] = gwh.A[0]; gwh.Wt[d] = gwh.Wt[0]; gwh.C[d] = gwh.C[0];
  }

  for (int s = 0; s < T_; ++s) {
    k_wmma_gemm<<<dim3((2 * H_) / 512, (B_ / 32) * KSLICES, 2), blk, 0, stream>>>(
        gzr, B_, 2 * H_, H_, KSLICES, PS_ZR);
    k_gates<<<dim3(B_, 2), blk, 0, stream>>>(HB, HZR, ZB, RH, sp, s);
    k_wmma_gemm<<<dim3(H_ / 512, (B_ / 32) * KSLICES, 2), blk, 0, stream>>>(
        gwh, B_, H_, H_, KSLICES, PS_HH);
    k_update<<<dim3(B_, 2), blk, 0, stream>>>(HB, RH, HH, ZB, (float*)d_out, sp, s);
  }

  k_proj_out<<<dim3(T_ * B_), blk, 0, stream>>>((float*)d_out);
  k_tail<<<dim3((2 * B_ * H_ / 4) / 256), blk, 0, stream>>>(HB, (float*)d_out);
}